// DGCNN_ECModule_90185723281830
// MI455X (gfx1250) — compile-verified
//
#include <hip/hip_runtime.h>
#include <hip/hip_bf16.h>

// ---------------------------------------------------------------------------
// DGCNN EdgeConv (knn -> edge feature -> 1x1 conv -> GroupNorm -> LeakyReLU ->
// max over K) for MI455X (gfx1250, wave32, WMMA).
//
// B=8, N=4096, C=3, O=64, K=80, GROUPS=2.
//
// k1: distances via V_WMMA_F32_16X16X4_F32 (A = 16 rows x 4(C pad), B = x^T),
//     keys kept entirely in LDS (16 x 4096 x 4B = 256KB of the 320KB WGP LDS),
//     exact per-row radix-select (4x8-bit LDS histogram) + deterministic
//     index-ordered compaction of the 80 nearest.
// k2: edge GEMM via two chained 16x16x4 WMMAs (channels 0..3 and 4..5),
//     fused max/min over K and group sums (monotonicity of LeakyReLU +
//     affine GroupNorm lets max over K commute with the normalization).
// ---------------------------------------------------------------------------

typedef float v2f __attribute__((ext_vector_type(2)));
typedef float v8f __attribute__((ext_vector_type(8)));

#define BB   8
#define NN   4096
#define OO   64
#define KNB  80
#define EPSF 1e-5
#define SLP  0.2f

// ---------------------------------------------------------------- k0: |x|^2
__global__ __launch_bounds__(256) void k0_x2(const float* __restrict__ x,
                                             float* __restrict__ x2) {
  int i = blockIdx.x * 256 + threadIdx.x;
  if (i < BB * NN) {
    const float* p = x + (size_t)i * 3;
    x2[i] = p[0] * p[0] + p[1] * p[1] + p[2] * p[2];
  }
}

// ---------------------------------------------------- k1: KNN (WMMA + radix)
// grid: B * N/16 blocks, 256 threads (8 waves). Dynamic LDS:
//   s_keys[16][4096] (sortable u32 distance keys) + s_x2[4096].
__global__ __launch_bounds__(256) void k1_knn(const float* __restrict__ xg,
                                              const float* __restrict__ x2g,
                                              int* __restrict__ idxg) {
  extern __shared__ unsigned char smem[];
  unsigned* s_keys = (unsigned*)smem;                       // 16*4096 u32
  float* s_x2 = (float*)(smem + 16 * 4096 * sizeof(unsigned));
  __shared__ unsigned s_hist[256], s_sL[256], s_sE[256], s_sel[2];

  const int tid = threadIdx.x, lane = tid & 31, wave = tid >> 5, lm = lane & 15;
  const int b = blockIdx.x >> 8;
  const int row0 = (blockIdx.x & 255) << 4;

  for (int i = tid; i < NN; i += 256) s_x2[i] = x2g[(size_t)b * NN + i];

  // A fragment: 16x4 f32 layout -> lanes 0-15: {K0,K1}; lanes 16-31: {K2,K3=0}
  const float* xr = xg + ((size_t)b * NN + row0 + lm) * 3;
  v2f afrag;
  afrag.x = (lane < 16) ? xr[0] : xr[2];
  afrag.y = (lane < 16) ? xr[1] : 0.0f;
  __syncthreads();

  v8f zero = {};
  for (int t = wave; t < NN / 16; t += 8) {
    int col = (t << 4) + lm;
    const float* xc = xg + ((size_t)b * NN + col) * 3;
    // B fragment: 4x16 f32 -> v0: K0 (lanes 0-15)/K2 (16-31); v1: K1/K3
    v2f bfrag;
    bfrag.x = (lane < 16) ? xc[0] : xc[2];
    bfrag.y = (lane < 16) ? xc[1] : 0.0f;
    v8f d = __builtin_amdgcn_wmma_f32_16x16x4_f32(false, afrag, false, bfrag,
                                                  (short)0, zero, false, false);
    // key = |x_m|^2 - 2<x_n,x_m>  (row term constant per row; drop it)
    float x2c = s_x2[col];
    int mb = (lane < 16) ? 0 : 8;
#pragma unroll
    for (int v = 0; v < 8; ++v) {
      float key = x2c - 2.0f * d[v];
      unsigned u = __float_as_uint(key);
      u = (u & 0x80000000u) ? ~u : (u | 0x80000000u);  // ascending-sortable
      s_keys[(unsigned)((mb + v) << 12) + (unsigned)col] = u;
    }
  }
  __syncthreads();

  // Exact top-80-smallest per row: 4-pass radix select + stable compaction.
  for (int r = 0; r < 16; ++r) {
    unsigned* keys = s_keys + (r << 12);
    if (tid == 0) { s_sel[0] = KNB; s_sel[1] = 0u; }
    __syncthreads();
    for (int shift = 24; shift >= 0; shift -= 8) {
      s_hist[tid] = 0;
      __syncthreads();
      unsigned pre = s_sel[1];
      unsigned hi = (shift == 24) ? 0u : (0xFFFFFFFFu << (shift + 8));
      for (int e = 0; e < 16; ++e) {
        unsigned u = keys[(tid << 4) + e];
        if ((u & hi) == pre) atomicAdd(&s_hist[(u >> shift) & 255u], 1u);
      }
      __syncthreads();
      if (tid == 0) {
        unsigned need = s_sel[0], cum = 0;
        for (int bin = 0; bin < 256; ++bin) {
          unsigned c = s_hist[bin];
          if (cum + c >= need) {
            s_sel[1] = pre | ((unsigned)bin << shift);
            s_sel[0] = need - cum;
            break;
          }
          cum += c;
        }
      }
      __syncthreads();
    }
    unsigned thr = s_sel[1], needE = s_sel[0];
    unsigned cL = 0, cE = 0;
    for (int e = 0; e < 16; ++e) {
      unsigned u = keys[(tid << 4) + e];
      cL += (u < thr);
      cE += (u == thr);
    }
    s_sL[tid] = cL;
    s_sE[tid] = cE;
    __syncthreads();
    for (int off = 1; off < 256; off <<= 1) {  // Hillis-Steele inclusive scan
      unsigned vL = (tid >= off) ? s_sL[tid - off] : 0u;
      unsigned vE = (tid >= off) ? s_sE[tid - off] : 0u;
      __syncthreads();
      s_sL[tid] += vL;
      s_sE[tid] += vE;
      __syncthreads();
    }
    unsigned totL = s_sL[255];
    unsigned pL = s_sL[tid] - cL, pE = s_sE[tid] - cE;
    int* op = idxg + ((size_t)b * NN + row0 + r) * (size_t)KNB;
    for (int e = 0; e < 16; ++e) {
      int col = (tid << 4) + e;
      unsigned u = keys[col];
      if (u < thr) {
        op[pL++] = col;
      } else if (u == thr) {
        if (pE < needE) op[totL + pE] = col;
        pE++;
      }
    }
    __syncthreads();
  }
}

// --------------------------------------- k2: edge conv (WMMA) + reductions
// grid: B*N/8 blocks, 256 threads; wave w handles point nglob = blk*8+w.
__global__ __launch_bounds__(256) void k2_conv(const float* __restrict__ xg,
                                               const float* __restrict__ wg_,
                                               const int* __restrict__ idxg,
                                               float* __restrict__ maxh,
                                               float* __restrict__ minh,
                                               float* __restrict__ partial) {
  __shared__ float s_W[OO * 6];
  __shared__ float s_gsum[8][4];
  const int tid = threadIdx.x, lane = tid & 31, w = tid >> 5, lm = lane & 15;
  for (int i = tid; i < OO * 6; i += 256) s_W[i] = wg_[i];

  const int nglob = blockIdx.x * 8 + w;
  const int b = nglob >> 12;  // N = 4096
  const float* cp = xg + (size_t)nglob * 3;
  const float c0 = cp[0], c1 = cp[1], c2 = cp[2];
  __syncthreads();

  // Hoisted B fragments (W^T), 4 col tiles x 2 K-chunks. Channels:
  // 0..2 = nbr-center, 3..5 = center; chunk0 = ch0..3, chunk1 = ch4..5(+pad).
  v2f bw0[4], bw1[4];
#pragma unroll
  for (int ct = 0; ct < 4; ++ct) {
    const float* wr = s_W + ((ct << 4) + lm) * 6;
    if (lane < 16) {
      bw0[ct].x = wr[0]; bw0[ct].y = wr[1];
      bw1[ct].x = wr[4]; bw1[ct].y = wr[5];
    } else {
      bw0[ct].x = wr[2]; bw0[ct].y = wr[3];
      bw1[ct].x = 0.0f;  bw1[ct].y = 0.0f;
    }
  }

  float mx[4], mn[4], sm[4], sq[4];
#pragma unroll
  for (int ct = 0; ct < 4; ++ct) {
    mx[ct] = -__builtin_huge_valf();
    mn[ct] = __builtin_huge_valf();
    sm[ct] = 0.0f;
    sq[ct] = 0.0f;
  }

  const int* ip = idxg + (size_t)nglob * KNB;
  v8f zero = {};
  for (int kt = 0; kt < KNB / 16; ++kt) {  // exactly 5 tiles of 16
    int nb = ip[(kt << 4) + lm];
    const float* np = xg + ((size_t)b * NN + nb) * 3;
    float e0 = np[0], e1 = np[1], e2 = np[2];
    v2f a0, a1;
    if (lane < 16) {
      a0.x = e0 - c0; a0.y = e1 - c1;   // ch0, ch1
      a1.x = c1;      a1.y = c2;        // ch4, ch5
    } else {
      a0.x = e2 - c2; a0.y = c0;        // ch2, ch3
      a1.x = 0.0f;    a1.y = 0.0f;      // ch6, ch7 pad
    }
#pragma unroll
    for (int ct = 0; ct < 4; ++ct) {
      v8f acc = __builtin_amdgcn_wmma_f32_16x16x4_f32(
          false, a0, false, bw0[ct], (short)0, zero, false, false);
      acc = __builtin_amdgcn_wmma_f32_16x16x4_f32(
          false, a1, false, bw1[ct], (short)0, acc, false, false);
#pragma unroll
      for (int v = 0; v < 8; ++v) {
        float h = acc[v];
        mx[ct] = fmaxf(mx[ct], h);
        mn[ct] = fminf(mn[ct], h);
        sm[ct] += h;
        sq[ct] += h * h;
      }
    }
  }

  float gs0 = 0, gs1 = 0, gq0 = 0, gq1 = 0;
  size_t obase = (size_t)nglob << 6;
#pragma unroll
  for (int ct = 0; ct < 4; ++ct) {
    float m2 = fmaxf(mx[ct], __shfl_xor(mx[ct], 16, 32));
    float l2 = fminf(mn[ct], __shfl_xor(mn[ct], 16, 32));
    float s2 = sm[ct] + __shfl_xor(sm[ct], 16, 32);
    float q2 = sq[ct] + __shfl_xor(sq[ct], 16, 32);
    if (lane < 16) {
      maxh[obase + (ct << 4) + lane] = m2;
      minh[obase + (ct << 4) + lane] = l2;
    }
#pragma unroll
    for (int off = 8; off >= 1; off >>= 1) {
      s2 += __shfl_xor(s2, off, 32);
      q2 += __shfl_xor(q2, off, 32);
    }
    if (ct < 2) { gs0 += s2; gq0 += q2; } else { gs1 += s2; gq1 += q2; }
  }
  if (lane == 0) {
    s_gsum[w][0] = gs0; s_gsum[w][1] = gs1;
    s_gsum[w][2] = gq0; s_gsum[w][3] = gq1;
  }
  __syncthreads();
  if (tid < 4) {
    float p = 0;
    for (int i = 0; i < 8; ++i) p += s_gsum[i][tid];
    partial[(size_t)blockIdx.x * 4 + tid] = p;  // deterministic
  }
}

// -------------------------------------------- k3: stats -> scale/shift per o
__global__ __launch_bounds__(512) void k3_stats(const float* __restrict__ partial,
                                                const float* __restrict__ gnw,
                                                const float* __restrict__ gnb,
                                                float* __restrict__ ss) {
  __shared__ float s_mean[16], s_inv[16];
  int t = threadIdx.x;
  if (t < 16) {
    int b = t >> 1, g = t & 1;
    double S = 0, Q = 0;
    for (int f = 0; f < 512; ++f) {
      size_t base = (size_t)(b * 512 + f) * 4;
      S += partial[base + g];
      Q += partial[base + 2 + g];
    }
    double cnt = (double)NN * KNB * (OO / 2);
    double mean = S / cnt;
    double var = Q / cnt - mean * mean;
    s_mean[t] = (float)mean;
    s_inv[t] = (float)rsqrt(var + EPSF);
  }
  __syncthreads();
  if (t < BB * OO) {
    int b = t >> 6, o = t & 63, g = o >> 5;
    float sc = gnw[o] * s_inv[b * 2 + g];
    float sh = gnb[o] - s_mean[b * 2 + g] * sc;
    ss[t * 2] = sc;
    ss[t * 2 + 1] = sh;
  }
}

// ------------------------------------------------------------- k4: epilogue
// max_k LeakyReLU(scale*h+shift) = LeakyReLU(scale*M+shift),
// M = max_k h if scale>=0 else min_k h (LeakyReLU monotone nondecreasing).
__global__ __launch_bounds__(256) void k4_out(const float* __restrict__ maxh,
                                              const float* __restrict__ minh,
                                              const float* __restrict__ ss,
                                              float* __restrict__ out) {
  int i = blockIdx.x * 256 + threadIdx.x;
  if (i < BB * NN * OO) {
    int o = i & 63;
    int b = i >> 18;  // / (N*O) = / 262144
    float sc = ss[(b * OO + o) * 2];
    float sh = ss[(b * OO + o) * 2 + 1];
    float m = (sc >= 0.0f) ? maxh[i] : minh[i];
    float v = sc * m + sh;
    out[i] = (v >= 0.0f) ? v : SLP * v;
  }
}

// ---------------------------------------------------------------------------
extern "C" void kernel_launch(void* const* d_in, const int* in_sizes, int n_in,
                              void* d_out, int out_size, void* d_ws, size_t ws_size,
                              hipStream_t stream) {
  const float* x = (const float*)d_in[0];       // (B,N,3)
  const float* conv_w = (const float*)d_in[1];  // (64,6)
  const float* gn_w = (const float*)d_in[2];    // (64)
  const float* gn_b = (const float*)d_in[3];    // (64)
  float* out = (float*)d_out;                   // (B,N,64)

  char* ws = (char*)d_ws;
  size_t oIdx = 0;                                      // B*N*K int32
  size_t oX2 = oIdx + (size_t)BB * NN * KNB * 4;        // B*N f32
  size_t oMax = oX2 + (size_t)BB * NN * 4;              // B*N*64 f32
  size_t oMin = oMax + (size_t)BB * NN * OO * 4;        // B*N*64 f32
  size_t oPart = oMin + (size_t)BB * NN * OO * 4;       // (B*N/8)*4 f32
  size_t oSS = oPart + (size_t)(BB * NN / 8) * 4 * 4;   // B*64*2 f32

  int* idx = (int*)(ws + oIdx);
  float* x2 = (float*)(ws + oX2);
  float* mx = (float*)(ws + oMax);
  float* mn = (float*)(ws + oMin);
  float* part = (float*)(ws + oPart);
  float* ssp = (float*)(ws + oSS);

  k0_x2<<<(BB * NN + 255) / 256, 256, 0, stream>>>(x, x2);

  const size_t smem1 = (size_t)16 * NN * 4 + (size_t)NN * 4;  // 278528 B
  k1_knn<<<BB * (NN / 16), 256, smem1, stream>>>(x, x2, idx);

  k2_conv<<<BB * NN / 8, 256, 0, stream>>>(x, conv_w, idx, mx, mn, part);

  k3_stats<<<1, 512, 0, stream>>>(part, gn_w, gn_b, ssp);

  k4_out<<<(BB * NN * OO + 255) / 256, 256, 0, stream>>>(mx, mn, ssp, out);
}